// PagedAttention_59158879535499
// MI455X (gfx1250) — compile-verified
//
#include <hip/hip_runtime.h>

#define B_ 32
#define H_ 16
#define D_ 128
#define BS_ 16
#define MAXB_ 128
#define SPLITS 8
#define LDK 132                 // padded LDS row stride (floats) -> conflict-free
#define SCALE_ 0.08838834764831845f   // 1/sqrt(128)

typedef float v2f __attribute__((ext_vector_type(2)));
typedef float v4f __attribute__((ext_vector_type(4)));
typedef float v8f __attribute__((ext_vector_type(8)));
typedef int   v4i __attribute__((ext_vector_type(4)));

typedef __attribute__((address_space(1))) v4i gv4i;   // global 128-bit
typedef __attribute__((address_space(3))) v4i lv4i;   // LDS 128-bit

#if defined(__has_builtin)
#if __has_builtin(__builtin_amdgcn_global_load_async_to_lds_b128)
#define HAVE_ASYNC 1
#else
#define HAVE_ASYNC 0
#endif
#else
#define HAVE_ASYNC 0
#endif

__device__ __forceinline__ void cp16(const float* g, float* l) {
#if HAVE_ASYNC
  __builtin_amdgcn_global_load_async_to_lds_b128((gv4i*)g, (lv4i*)l, 0, 0);
#else
  *(v4f*)l = *(const v4f*)g;
#endif
}

__device__ __forceinline__ void wait_cp() {
#if HAVE_ASYNC
#if __has_builtin(__builtin_amdgcn_s_wait_asynccnt)
  __builtin_amdgcn_s_wait_asynccnt(0);
#else
  asm volatile("s_wait_asynccnt 0x0" ::: "memory");
#endif
#endif
}

__device__ __forceinline__ int imin(int a, int b) { return a < b ? a : b; }

// ---------------------------------------------------------------------------
// Kernel 1: one wave (32 threads) per (head, split, batch).
// Scores: C[m=token][n=replicated] = K_tile(16x4chunk) x q_bcast(4x16), 32 WMMAs.
// PV:     C[m=dim][n=replicated]  = V^T_tile(16x4tok)  x p_bcast(4x16), 32 WMMAs.
// ---------------------------------------------------------------------------
__global__ __launch_bounds__(32)
void pa_split_kernel(const float* __restrict__ q, const float* __restrict__ kc,
                     const float* __restrict__ vc, const int* __restrict__ bt,
                     const int* __restrict__ cl, float* __restrict__ wm,
                     float* __restrict__ wl, float* __restrict__ wacc) {
  __shared__ float sm[4 * BS_ * LDK + D_];  // K dbl-buf, V dbl-buf, q
  float* kbuf = sm;
  float* vbuf = sm + 2 * BS_ * LDK;
  float* qs = sm + 4 * BS_ * LDK;

  const int h = blockIdx.x;
  const int sp = blockIdx.y;
  const int b = blockIdx.z;
  const int lane = threadIdx.x & 31;
  const int tloc = lane & 15;  // row (token or dim) within tile
  const int hi = lane >> 4;    // lane half
  const int dlt = hi * 2;      // K-offset for high half per WMMA layout

  const int ctx = cl[b];
  const int nbt = (ctx + BS_ - 1) / BS_;
  const int per = (nbt + SPLITS - 1) / SPLITS;
  const int bs0 = sp * per;
  const int bs1 = imin(bs0 + per, nbt);
  const int widx = (b * H_ + h) * SPLITS + sp;

  if (bs0 >= bs1) {  // empty split: neutral partial
    if (lane == 0) { wm[widx] = -1e30f; wl[widx] = 0.f; }
    for (int d = lane; d < D_; d += 32) wacc[(size_t)widx * D_ + d] = 0.f;
    return;
  }

  // q[h][0..127] -> LDS (one b128 per lane)
  *(v4f*)(qs + lane * 4) = *(const v4f*)(q + ((size_t)b * H_ + h) * D_ + lane * 4);

  auto prefetch = [&](int blkidx, int buf) {
    const int pb = bt[b * MAXB_ + blkidx];
    const float* kg = kc + ((size_t)pb * BS_ * H_ + h) * D_;
    const float* vg = vc + ((size_t)pb * BS_ * H_ + h) * D_;
#pragma unroll
    for (int t = 0; t < BS_; ++t) {  // 512B coalesced per token row
      cp16(kg + (size_t)t * H_ * D_ + lane * 4, kbuf + buf * BS_ * LDK + t * LDK + lane * 4);
      cp16(vg + (size_t)t * H_ * D_ + lane * 4, vbuf + buf * BS_ * LDK + t * LDK + lane * 4);
    }
  };

  prefetch(bs0, 0);

  float run_m = -1e30f, run_l = 0.f;
  const v8f vzero = {0.f, 0.f, 0.f, 0.f, 0.f, 0.f, 0.f, 0.f};
  v8f acc[8];
#pragma unroll
  for (int i = 0; i < 8; ++i) acc[i] = vzero;

  for (int ib = bs0; ib < bs1; ++ib) {
    const int buf = (ib - bs0) & 1;
    wait_cp();                                   // buffer `buf` is resident
    if (ib + 1 < bs1) prefetch(ib + 1, buf ^ 1); // overlap next block's DMA

    const float* kb = kbuf + buf * BS_ * LDK;
    const float* vb = vbuf + buf * BS_ * LDK;

    // ---- QK^T: 32x WMMA f32 16x16x4, two accumulators for ILP ----
    v8f ce = vzero, co = vzero;
#pragma unroll
    for (int c = 0; c < 32; c += 2) {
      v2f a0 = *(const v2f*)(kb + tloc * LDK + 4 * c + dlt);
      v2f b0 = *(const v2f*)(qs + 4 * c + dlt);
      ce = __builtin_amdgcn_wmma_f32_16x16x4_f32(false, a0, false, b0, (short)0, ce, false, false);
      v2f a1 = *(const v2f*)(kb + tloc * LDK + 4 * (c + 1) + dlt);
      v2f b1 = *(const v2f*)(qs + 4 * (c + 1) + dlt);
      co = __builtin_amdgcn_wmma_f32_16x16x4_f32(false, a1, false, b1, (short)0, co, false, false);
    }

    // ---- online softmax (stats are wave-uniform: one head per wave) ----
    float s[8], p[8];
    float bm = -1e30f;
    const int tbase = ib * BS_ + hi * 8;
#pragma unroll
    for (int r = 0; r < 8; ++r) {
      float sv = (ce[r] + co[r]) * SCALE_;
      sv = (tbase + r < ctx) ? sv : -1e30f;
      s[r] = sv;
      bm = fmaxf(bm, sv);
    }
    bm = fmaxf(bm, __shfl_xor(bm, 16));
    const float mnew = fmaxf(run_m, bm);
    const float f = __expf(run_m - mnew);
    float bsum = 0.f;
#pragma unroll
    for (int r = 0; r < 8; ++r) { p[r] = __expf(s[r] - mnew); bsum += p[r]; }
    bsum += __shfl_xor(bsum, 16);
    run_l = run_l * f + bsum;
    run_m = mnew;
#pragma unroll
    for (int i = 0; i < 8; ++i) acc[i] = acc[i] * f;

    // build p broadcast B-operands (cross-half moves)
    float xp[8];
#pragma unroll
    for (int r = 0; r < 8; ++r) xp[r] = __shfl_xor(p[r], 16);
    const bool lo = (hi == 0);
    float bx[4], by[4];
    bx[0] = lo ? p[0] : xp[2];  by[0] = lo ? p[1] : xp[3];
    bx[1] = lo ? p[4] : xp[6];  by[1] = lo ? p[5] : xp[7];
    bx[2] = lo ? xp[0] : p[2];  by[2] = lo ? xp[1] : p[3];
    bx[3] = lo ? xp[4] : p[6];  by[3] = lo ? xp[5] : p[7];

    // ---- PV: 32x WMMA f32 16x16x4, out[d] accumulated in 8 C tiles ----
#pragma unroll
    for (int dc = 0; dc < 8; ++dc) {
      const float* vcol = vb + 16 * dc + tloc;
#pragma unroll
      for (int j = 0; j < 4; ++j) {
        v2f a; a.x = vcol[(4 * j + dlt) * LDK]; a.y = vcol[(4 * j + 1 + dlt) * LDK];
        v2f bb; bb.x = bx[j]; bb.y = by[j];
        acc[dc] = __builtin_amdgcn_wmma_f32_16x16x4_f32(false, a, false, bb, (short)0, acc[dc], false, false);
      }
    }
  }

  if (lane == 0) { wm[widx] = run_m; wl[widx] = run_l; }
  float* ap = wacc + (size_t)widx * D_;
#pragma unroll
  for (int dc = 0; dc < 8; ++dc)
#pragma unroll
    for (int r = 0; r < 8; ++r)
      if (tloc == r) ap[16 * dc + r + hi * 8] = acc[dc][r];
}

// ---------------------------------------------------------------------------
// Kernel 2: combine SPLITS partials per (b,h); one thread per output dim.
// ---------------------------------------------------------------------------
__global__ __launch_bounds__(128)
void pa_combine_kernel(const float* __restrict__ wm, const float* __restrict__ wl,
                       const float* __restrict__ wacc, float* __restrict__ out) {
  const int bh = blockIdx.x;
  const int d = threadIdx.x;
  float M = -1e30f;
#pragma unroll
  for (int s = 0; s < SPLITS; ++s) M = fmaxf(M, wm[bh * SPLITS + s]);
  float L = 0.f, o = 0.f;
#pragma unroll
  for (int s = 0; s < SPLITS; ++s) {
    const float w = __expf(wm[bh * SPLITS + s] - M);
    L += w * wl[bh * SPLITS + s];
    o += w * wacc[(size_t)(bh * SPLITS + s) * D_ + d];
  }
  out[(size_t)bh * D_ + d] = o / L;
}

extern "C" void kernel_launch(void* const* d_in, const int* in_sizes, int n_in,
                              void* d_out, int out_size, void* d_ws, size_t ws_size,
                              hipStream_t stream) {
  const float* q  = (const float*)d_in[0];
  const float* kc = (const float*)d_in[1];
  const float* vc = (const float*)d_in[2];
  const int* bt   = (const int*)d_in[3];
  const int* cl   = (const int*)d_in[4];

  float* wm   = (float*)d_ws;                       // [B*H*SPLITS]
  float* wl   = wm + B_ * H_ * SPLITS;              // [B*H*SPLITS]
  float* wacc = wl + B_ * H_ * SPLITS;              // [B*H*SPLITS][D]

  dim3 g1(H_, SPLITS, B_);
  pa_split_kernel<<<g1, 32, 0, stream>>>(q, kc, vc, bt, cl, wm, wl, wacc);
  pa_combine_kernel<<<B_ * H_, D_, 0, stream>>>(wm, wl, wacc, (float*)d_out);
}